// Model_28226525069324
// MI455X (gfx1250) — compile-verified
//
#include <hip/hip_runtime.h>

#define N_NODES   100000
#define N_EDGES   1600000
#define N_GRAPHS  64
#define CH        128
#define HIDDEN_N  256
#define N_FEAT    16
#define N_TARGETS 8
#define N_LAYERS  4
#define BN_EPS    1e-5f

typedef __attribute__((ext_vector_type(2))) float v2f;
typedef __attribute__((ext_vector_type(8))) float v8f;

// ---------------- degree / CSR construction (once per call) ----------------

__global__ void k_count(const int* __restrict__ dst, int* __restrict__ cnt) {
  int e = blockIdx.x * blockDim.x + threadIdx.x;
  if (e < N_EDGES) atomicAdd(&cnt[dst[e]], 1);
}

__global__ void k_dinv(const int* __restrict__ cnt, float* __restrict__ dinv) {
  int i = blockIdx.x * blockDim.x + threadIdx.x;
  if (i < N_NODES) dinv[i] = rsqrtf((float)cnt[i] + 1.0f);  // deg includes self-loop
}

__global__ __launch_bounds__(1024) void k_scan(const int* __restrict__ cnt,
                                               int* __restrict__ row_ptr,
                                               int* __restrict__ wr_ptr) {
  __shared__ int s[1024];
  int tid = threadIdx.x;
  int running = 0;
  for (int base = 0; base < N_NODES; base += 1024) {
    int i = base + tid;
    int v = (i < N_NODES) ? cnt[i] : 0;
    s[tid] = v;
    __syncthreads();
    for (int off = 1; off < 1024; off <<= 1) {
      int t = (tid >= off) ? s[tid - off] : 0;
      __syncthreads();
      s[tid] += t;
      __syncthreads();
    }
    if (i < N_NODES) { int ex = running + s[tid] - v; row_ptr[i] = ex; wr_ptr[i] = ex; }
    int tot = s[1023];
    __syncthreads();
    running += tot;
  }
  if (tid == 0) row_ptr[N_NODES] = running;
}

__global__ void k_csr(const int* __restrict__ src, const int* __restrict__ dst,
                      const float* __restrict__ dinv, int* __restrict__ wr_ptr,
                      int* __restrict__ csr_src, float* __restrict__ csr_w) {
  int e = blockIdx.x * blockDim.x + threadIdx.x;
  if (e < N_EDGES) {
    int d = dst[e], s = src[e];
    int pos = atomicAdd(&wr_ptr[d], 1);
    csr_src[pos] = s;
    csr_w[pos] = dinv[s] * dinv[d];
  }
}

// ---------------- embedding gather ----------------

__global__ void k_embed(const int* __restrict__ nt, const float* __restrict__ emb,
                        float* __restrict__ X) {
  int i = blockIdx.x * blockDim.x + threadIdx.x;
  if (i < N_NODES * CH) {
    int node = i >> 7, c = i & 127;
    X[i] = emb[nt[node] * CH + c];
  }
}

// ---------------- BatchNorm statistics (channel-per-thread) ----------------

__global__ __launch_bounds__(128) void k_bn_stats(const float* __restrict__ X,
                                                  float* __restrict__ bsum,
                                                  float* __restrict__ bssq) {
  int c = threadIdx.x;
  float s = 0.f, ss = 0.f;
  for (int row = blockIdx.x; row < N_NODES; row += gridDim.x) {
    float v = X[row * CH + c];
    s += v; ss += v * v;
  }
  atomicAdd(&bsum[c], s);
  atomicAdd(&bssq[c], ss);
}

__global__ void k_bn_fin(const float* __restrict__ bsum, const float* __restrict__ bssq,
                         const float* __restrict__ gamma, const float* __restrict__ beta,
                         float* __restrict__ scale, float* __restrict__ shift) {
  int c = threadIdx.x;
  if (c < CH) {
    float mean = bsum[c] * (1.0f / N_NODES);
    float var  = bssq[c] * (1.0f / N_NODES) - mean * mean;  // biased, like jnp.var
    float sc   = gamma[c] * rsqrtf(var + BN_EPS);
    scale[c] = sc;
    shift[c] = beta[c] - mean * sc;
  }
}

// ---------------- fused BN + GEMM via V_WMMA_F32_16X16X4_F32 ----------------
// XW = (X*scale+shift) @ W   [N x 128] @ [128 x 128]
// 256 threads = 8 waves; wave w owns output columns [16w,16w+16).
// B strip (128x16) is loaded ONCE per block into 64 VGPRs/lane and reused
// across 16 M-tiles; only the A tile cycles through LDS (ds_load_2addr_b32).

#define M_TILES_PER_WG 16

__global__ __launch_bounds__(256) void k_gemm_bn(const float* __restrict__ X,
                                                 const float* __restrict__ W,
                                                 const float* __restrict__ scale,
                                                 const float* __restrict__ shift,
                                                 float* __restrict__ XW) {
  __shared__ float sA[16 * 129];   // padded row stride breaks bank conflicts
  int tid  = threadIdx.x;
  int wave = tid >> 5;
  int lane = tid & 31;
  int h = lane >> 4;   // half-wave select (K pair)
  int l = lane & 15;   // row / column index within tile
  int n0 = wave * 16;

  // B fragments, resident in registers for the whole block.
  // Layout per ISA 7.12.2: VGPR0 rows {K=4j | K=4j+2}, VGPR1 rows {K=4j+1 | K=4j+3}.
  v2f breg[32];
#pragma unroll
  for (int j = 0; j < 32; ++j) {
    int ka = 4 * j + 2 * h;
    breg[j].x = W[ka * CH + n0 + l];
    breg[j].y = W[(ka + 1) * CH + n0 + l];
  }

  int m_base = blockIdx.x * (M_TILES_PER_WG * 16);
  for (int t = 0; t < M_TILES_PER_WG; ++t) {
    int m0 = m_base + t * 16;              // N_NODES % 16 == 0, rows always valid
    if (m0 >= N_NODES) break;
    __syncthreads();                       // sA reuse hazard between tiles
    for (int i = tid; i < 16 * CH; i += 256) {
      int r = i >> 7, c = i & 127;
      sA[r * 129 + c] = X[(m0 + r) * CH + c] * scale[c] + shift[c];  // fused BN
    }
    __syncthreads();

    v8f acc = {};
#pragma unroll
    for (int j = 0; j < 32; ++j) {
      int ka = 4 * j + 2 * h;
      v2f a;
      // A 16x4 layout: lanes0-15 K={4j,4j+1}, lanes16-31 K={4j+2,4j+3}, M=l
      a.x = sA[l * 129 + ka];
      a.y = sA[l * 129 + ka + 1];
      acc = __builtin_amdgcn_wmma_f32_16x16x4_f32(false, a, false, breg[j],
                                                  (short)0, acc, false, false);
    }
#pragma unroll
    for (int v = 0; v < 8; ++v) {
      // D layout: VGPR v -> rows v (lanes0-15) and v+8 (lanes16-31), col = n0+l
      XW[(m0 + v + 8 * h) * CH + n0 + l] = acc[v];
    }
  }
}

// ---------------- CSR aggregation + bias + ReLU (+ residual) ----------------
// Block of 256 handles 2 nodes; 128 threads per node (one channel each).
// Row gathers are 512B contiguous -> served from L2 (x fits in 192MB L2).

__global__ __launch_bounds__(256) void k_agg(const float* __restrict__ XW,
                                             const float* __restrict__ bias,
                                             const float* __restrict__ dinv,
                                             const int* __restrict__ row_ptr,
                                             const int* __restrict__ csr_src,
                                             const float* __restrict__ csr_w,
                                             const float* __restrict__ residual,
                                             float* __restrict__ out) {
  int tid = threadIdx.x;
  int c = tid & 127;
  int node = blockIdx.x * 2 + (tid >> 7);
  if (node >= N_NODES) return;
  float di = dinv[node];
  float acc = XW[node * CH + c] * di * di;      // self-loop term
  int e0 = row_ptr[node], e1 = row_ptr[node + 1];
  for (int e = e0; e < e1; ++e)
    acc += XW[csr_src[e] * CH + c] * csr_w[e];
  acc += bias[c];
  acc = fmaxf(acc, 0.0f);
  if (residual) acc += residual[node * CH + c];
  out[node * CH + c] = acc;
}

// ---------------- per-graph mean pool ----------------

__global__ __launch_bounds__(128) void k_pool(const float* __restrict__ X,
                                              const int* __restrict__ batch,
                                              float* __restrict__ pooled,
                                              int* __restrict__ gcnt) {
  __shared__ float accs[N_GRAPHS * CH];
  __shared__ int lcnt[N_GRAPHS];
  int c = threadIdx.x;
  for (int g = 0; g < N_GRAPHS; ++g) accs[g * CH + c] = 0.0f;
  if (c < N_GRAPHS) lcnt[c] = 0;
  __syncthreads();
  for (int row = blockIdx.x; row < N_NODES; row += gridDim.x) {
    int g = batch[row];
    accs[g * CH + c] += X[row * CH + c];
    if (c == 0) lcnt[g] += 1;
  }
  __syncthreads();
  for (int g = 0; g < N_GRAPHS; ++g) atomicAdd(&pooled[g * CH + c], accs[g * CH + c]);
  if (c < N_GRAPHS) atomicAdd(&gcnt[c], lcnt[c]);
}

// ---------------- tiny MLP head: [64x144]@[144x256] relu, [64x256]@[256x8] ----------------

__global__ __launch_bounds__(256) void k_mlp(const float* __restrict__ pooled,
                                             const int* __restrict__ gcnt,
                                             const float* __restrict__ features,
                                             const float* __restrict__ hW,
                                             const float* __restrict__ hb,
                                             const float* __restrict__ fW,
                                             const float* __restrict__ fb,
                                             float* __restrict__ out) {
  __shared__ float P[N_GRAPHS * (CH + N_FEAT)];
  __shared__ float H[N_GRAPHS * HIDDEN_N];
  int tid = threadIdx.x;
  for (int i = tid; i < N_GRAPHS * (CH + N_FEAT); i += 256) {
    int g = i / (CH + N_FEAT), k = i % (CH + N_FEAT);
    P[i] = (k < CH) ? pooled[g * CH + k] / fmaxf((float)gcnt[g], 1.0f)
                    : features[g * N_FEAT + (k - CH)];
  }
  __syncthreads();
  for (int g = 0; g < N_GRAPHS; ++g) {        // thread owns hidden unit `tid`
    float s = hb[tid];
    for (int k = 0; k < CH + N_FEAT; ++k)
      s += P[g * (CH + N_FEAT) + k] * hW[k * HIDDEN_N + tid];
    H[g * HIDDEN_N + tid] = fmaxf(s, 0.0f);
  }
  __syncthreads();
  for (int i = tid; i < N_GRAPHS * N_TARGETS; i += 256) {
    int g = i / N_TARGETS, j = i % N_TARGETS;
    float s = fb[j];
    for (int k = 0; k < HIDDEN_N; ++k)
      s += H[g * HIDDEN_N + k] * fW[k * N_TARGETS + j];
    out[i] = s;
  }
}

// ---------------- orchestration ----------------

extern "C" void kernel_launch(void* const* d_in, const int* in_sizes, int n_in,
                              void* d_out, int out_size, void* d_ws, size_t ws_size,
                              hipStream_t stream) {
  (void)in_sizes; (void)n_in; (void)out_size; (void)ws_size;
  const int*   node_types = (const int*)d_in[0];
  const int*   ei         = (const int*)d_in[1];
  const int*   src        = ei;
  const int*   dst        = ei + N_EDGES;
  const float* features   = (const float*)d_in[2];
  const int*   batch      = (const int*)d_in[3];
  const float* embedding  = (const float*)d_in[4];
  const float* bn_gamma   = (const float*)d_in[5];   // [4][2][128]
  const float* bn_beta    = (const float*)d_in[6];
  const float* conv_W     = (const float*)d_in[7];   // [4][2][128][128]
  const float* conv_b     = (const float*)d_in[8];
  const float* hidden_W   = (const float*)d_in[9];
  const float* hidden_b   = (const float*)d_in[10];
  const float* fc_W       = (const float*)d_in[11];
  const float* fc_b       = (const float*)d_in[12];
  float* out = (float*)d_out;

  char* w = (char*)d_ws;
  float* X    = (float*)w; w += (size_t)N_NODES * CH * 4;
  float* MID  = (float*)w; w += (size_t)N_NODES * CH * 4;
  float* XW   = (float*)w; w += (size_t)N_NODES * CH * 4;
  float* dinv = (float*)w; w += (size_t)N_NODES * 4;
  int* cnt     = (int*)w;  w += (size_t)N_NODES * 4;
  int* row_ptr = (int*)w;  w += (size_t)(N_NODES + 1) * 4;
  int* wr_ptr  = (int*)w;  w += (size_t)N_NODES * 4;
  int* csr_src = (int*)w;  w += (size_t)N_EDGES * 4;
  float* csr_w = (float*)w; w += (size_t)N_EDGES * 4;
  float* bn_sum   = (float*)w; w += CH * 4;   // bn_sum + bn_ssq contiguous (one memset)
  float* bn_ssq   = (float*)w; w += CH * 4;
  float* bn_scale = (float*)w; w += CH * 4;
  float* bn_shift = (float*)w; w += CH * 4;
  float* pooled   = (float*)w; w += (size_t)N_GRAPHS * CH * 4;
  int* gcnt       = (int*)w;   w += (size_t)N_GRAPHS * 4;

  hipMemsetAsync(cnt, 0, (size_t)N_NODES * 4, stream);
  hipMemsetAsync(pooled, 0, (size_t)(N_GRAPHS * CH) * 4, stream);
  hipMemsetAsync(gcnt, 0, (size_t)N_GRAPHS * 4, stream);

  k_count<<<(N_EDGES + 255) / 256, 256, 0, stream>>>(dst, cnt);
  k_dinv<<<(N_NODES + 255) / 256, 256, 0, stream>>>(cnt, dinv);
  k_scan<<<1, 1024, 0, stream>>>(cnt, row_ptr, wr_ptr);
  k_csr<<<(N_EDGES + 255) / 256, 256, 0, stream>>>(src, dst, dinv, wr_ptr, csr_src, csr_w);
  k_embed<<<(N_NODES * CH + 255) / 256, 256, 0, stream>>>(node_types, embedding, X);

  const int gemm_grid = (N_NODES / 16 + M_TILES_PER_WG - 1) / M_TILES_PER_WG;  // 391
  for (int l = 0; l < N_LAYERS; ++l) {
    for (int s = 0; s < 2; ++s) {
      const float* in  = (s == 0) ? X : MID;
      float* outp      = (s == 0) ? MID : X;
      const float* res = (s == 0) ? (const float*)nullptr : X;
      int li = l * 2 + s;
      hipMemsetAsync(bn_sum, 0, 2 * CH * 4, stream);
      k_bn_stats<<<512, 128, 0, stream>>>(in, bn_sum, bn_ssq);
      k_bn_fin<<<1, 128, 0, stream>>>(bn_sum, bn_ssq, bn_gamma + li * CH,
                                      bn_beta + li * CH, bn_scale, bn_shift);
      k_gemm_bn<<<gemm_grid, 256, 0, stream>>>(in, conv_W + (size_t)li * CH * CH,
                                               bn_scale, bn_shift, XW);
      k_agg<<<(N_NODES + 1) / 2, 256, 0, stream>>>(XW, conv_b + li * CH, dinv,
                                                   row_ptr, csr_src, csr_w, res, outp);
    }
  }
  k_pool<<<240, 128, 0, stream>>>(X, batch, pooled, gcnt);
  k_mlp<<<1, 256, 0, stream>>>(pooled, gcnt, features, hidden_W, hidden_b, fc_W, fc_b, out);
}